// GCN256_36816459662020
// MI455X (gfx1250) — compile-verified
//
#include <hip/hip_runtime.h>
#include <hip/hip_bf16.h>
#include <stdint.h>

// ---------------------------------------------------------------------------
// GCN forward for MI455X (gfx1250, wave32).
//   - Dense GEMMs via V_WMMA_F32_16X16X4_F32 (fp32 precision == reference).
//   - Edge aggregation via CSR-by-dst gather (built once per launch) instead
//     of 330M hot-path f32 atomics; fused self-loop+bias+ReLU+BN+residual.
//   - Whole working set (~45 MB) is L2-resident (192 MB L2, 23.3 TB/s HBM
//     never touched in steady state).
// ---------------------------------------------------------------------------

#define N_NODES   20000
#define NODE_DIMC 128
#define HDIM      256
#define LAYERS    4
#define NGRAPH    128
#define EPSC      1e-5f
#define SCAN_T    1024

typedef __attribute__((ext_vector_type(2))) float v2f;
typedef __attribute__((ext_vector_type(8))) float v8f;

// ---- in-degree count over dst (self loop accounted as +1 later) ----
__global__ void deg_kernel(const int* __restrict__ dst, int* __restrict__ cnt, int E) {
    int e = blockIdx.x * blockDim.x + threadIdx.x;
    if (e < E) atomicAdd(&cnt[dst[e]], 1);
}

__global__ void dinv_kernel(const int* __restrict__ cnt, float* __restrict__ dinv, int n) {
    int i = blockIdx.x * blockDim.x + threadIdx.x;
    if (i < n) dinv[i] = rsqrtf((float)(cnt[i] + 1));  // +1 = self loop
}

// ---- single-block exclusive scan of cnt[0..n) -> rowptr[0..n] ----
__global__ void scan_kernel(const int* __restrict__ cnt, int* __restrict__ rowptr, int n) {
    __shared__ int part[SCAN_T];
    const int t  = threadIdx.x;
    const int CH = (n + SCAN_T - 1) / SCAN_T;
    const int base = t * CH;
    int s = 0;
    for (int i = 0; i < CH; ++i) {
        int idx = base + i;
        if (idx < n) s += cnt[idx];
    }
    part[t] = s;
    __syncthreads();
    // inclusive Hillis-Steele scan over partials
    for (int off = 1; off < SCAN_T; off <<= 1) {
        int v = (t >= off) ? part[t - off] : 0;
        __syncthreads();
        part[t] += v;
        __syncthreads();
    }
    int run = (t == 0) ? 0 : part[t - 1];  // exclusive prefix for this chunk
    for (int i = 0; i < CH; ++i) {
        int idx = base + i;
        if (idx < n) { rowptr[idx] = run; run += cnt[idx]; }
    }
    if (t == SCAN_T - 1) rowptr[n] = part[SCAN_T - 1];  // total = E
}

// ---- bin edges by dst: col[pos]=src, wght[pos]=dinv[src]*dinv[dst] ----
__global__ void fill_kernel(const int* __restrict__ src, const int* __restrict__ dst,
                            const float* __restrict__ dinv, int* __restrict__ cursor,
                            int* __restrict__ col, float* __restrict__ wght, int E) {
    int e = blockIdx.x * blockDim.x + threadIdx.x;
    if (e >= E) return;
    const int s = src[e], d = dst[e];
    const int pos = atomicAdd(&cursor[d], 1);
    col[pos]  = s;
    wght[pos] = dinv[s] * dinv[d];
}

// ---- fp32 WMMA GEMM: C[M x Nc] = A[M x K] @ W[K x Nc] (+ bias) ----
// One wave per 16x16 output tile; K stepped by 4 with v_wmma_f32_16x16x4_f32.
// A 16x4 : lanes 0-15 hold {K=k+0,k+1}, lanes 16-31 hold {K=k+2,k+3},
//          row = tile_m + (lane & 15).
// B 4x16 : VGPR0 = rows k+0 / k+2, VGPR1 = rows k+1 / k+3, col = tn+(lane&15).
// D 16x16: VGPR r = row tile_m + r + 8*(lane>>4), col tn + (lane&15).
template <int K, bool BIAS>
__global__ void wmma_gemm_kernel(const float* __restrict__ A,
                                 const float* __restrict__ W,
                                 const float* __restrict__ bias,
                                 float* __restrict__ C,
                                 int M, int Nc) {
    const int wave   = (blockIdx.x * blockDim.x + threadIdx.x) >> 5;
    const int lane   = threadIdx.x & 31;
    const int tilesN = Nc >> 4;
    const int tm = (wave / tilesN) << 4;
    const int tn = (wave % tilesN) << 4;
    if (tm >= M) return;                 // wave-uniform (EXEC stays all-1 for WMMA)
    const int half = lane >> 4;
    const int lm   = lane & 15;

    v8f acc = {};
    const float* arow = A + (size_t)(tm + lm) * K + 2 * half;
    const float* bcol = W + tn + lm;

#pragma unroll 8
    for (int k = 0; k < K; k += 4) {
        v2f a;
        {
            const float2 t = *(const float2*)(arow + k);
            a[0] = t.x; a[1] = t.y;
        }
        v2f b;
        b[0] = bcol[(size_t)(k + 2 * half)     * Nc];
        b[1] = bcol[(size_t)(k + 2 * half + 1) * Nc];
        acc = __builtin_amdgcn_wmma_f32_16x16x4_f32(
            false, a, false, b, (short)0, acc, false, false);
    }

    const float bv = BIAS ? bias[tn + lm] : 0.0f;
#pragma unroll
    for (int r = 0; r < 8; ++r) {
        const int row = tm + r + 8 * half;
        C[(size_t)row * Nc + tn + lm] = acc[r] + bv;
    }
}

// ---- fused CSR gather + self loop + bias + ReLU + BN + residual ----
// One wave per node; lane covers cols [4*lane..4*lane+3] and [128+4*lane..].
__global__ void aggregate_kernel(const float* __restrict__ m,
                                 const int* __restrict__ rowptr,
                                 const int* __restrict__ col,
                                 const float* __restrict__ wght,
                                 const float* __restrict__ dinv,
                                 const float* __restrict__ cb,
                                 const float* __restrict__ gam,
                                 const float* __restrict__ bet,
                                 const float* __restrict__ mu,
                                 const float* __restrict__ var,
                                 float* __restrict__ h) {
    const int node = (blockIdx.x * blockDim.x + threadIdx.x) >> 5;
    const int lane = threadIdx.x & 31;
    if (node >= N_NODES) return;
    const int beg = rowptr[node];
    const int end = rowptr[node + 1];

    float a0x = 0.f, a0y = 0.f, a0z = 0.f, a0w = 0.f;
    float a1x = 0.f, a1y = 0.f, a1z = 0.f, a1w = 0.f;

    for (int k = beg; k < end; ++k) {            // wave-uniform loop (avg deg ~16)
        const int   s = col[k];
        const float w = wght[k];
        const float4* row = (const float4*)(m + (size_t)s * HDIM);
        const float4 v0 = row[lane];             // cols 4*lane .. +3
        const float4 v1 = row[32 + lane];        // cols 128+4*lane .. +3
        a0x = fmaf(v0.x, w, a0x); a0y = fmaf(v0.y, w, a0y);
        a0z = fmaf(v0.z, w, a0z); a0w = fmaf(v0.w, w, a0w);
        a1x = fmaf(v1.x, w, a1x); a1y = fmaf(v1.y, w, a1y);
        a1z = fmaf(v1.z, w, a1z); a1w = fmaf(v1.w, w, a1w);
    }

    // self loop: norm = dinv[node]^2
    {
        const float dv = dinv[node];
        const float sw = dv * dv;
        const float4* row = (const float4*)(m + (size_t)node * HDIM);
        const float4 v0 = row[lane];
        const float4 v1 = row[32 + lane];
        a0x = fmaf(v0.x, sw, a0x); a0y = fmaf(v0.y, sw, a0y);
        a0z = fmaf(v0.z, sw, a0z); a0w = fmaf(v0.w, sw, a0w);
        a1x = fmaf(v1.x, sw, a1x); a1y = fmaf(v1.y, sw, a1y);
        a1z = fmaf(v1.z, sw, a1z); a1w = fmaf(v1.w, sw, a1w);
    }

    // epilogue: +bias, ReLU, BN(eval), residual into h — per-column params
    const float4* cb4  = (const float4*)cb;
    const float4* gm4  = (const float4*)gam;
    const float4* bt4  = (const float4*)bet;
    const float4* mu4  = (const float4*)mu;
    const float4* vr4  = (const float4*)var;
    float4* hrow = (float4*)(h + (size_t)node * HDIM);

#pragma unroll
    for (int seg = 0; seg < 2; ++seg) {
        const int ci = seg * 32 + lane;
        const float4 b4 = cb4[ci], g4 = gm4[ci], e4 = bt4[ci], m4 = mu4[ci], s4 = vr4[ci];
        float vx = seg ? a1x : a0x, vy = seg ? a1y : a0y;
        float vz = seg ? a1z : a0z, vw = seg ? a1w : a0w;
        vx = fmaxf(vx + b4.x, 0.f); vy = fmaxf(vy + b4.y, 0.f);
        vz = fmaxf(vz + b4.z, 0.f); vw = fmaxf(vw + b4.w, 0.f);
        vx = (vx - m4.x) * rsqrtf(s4.x + EPSC) * g4.x + e4.x;
        vy = (vy - m4.y) * rsqrtf(s4.y + EPSC) * g4.y + e4.y;
        vz = (vz - m4.z) * rsqrtf(s4.z + EPSC) * g4.z + e4.z;
        vw = (vw - m4.w) * rsqrtf(s4.w + EPSC) * g4.w + e4.w;
        float4 hv = hrow[ci];
        hv.x += vx; hv.y += vy; hv.z += vz; hv.w += vw;
        hrow[ci] = hv;
    }
}

// ---- per-graph segment sums + counts (one-time, ~5M cheap atomics) ----
__global__ void pool_kernel(const float* __restrict__ h,
                            const int* __restrict__ batch,
                            float* __restrict__ sums,
                            float* __restrict__ counts) {
    const int i = blockIdx.x;
    const int j = threadIdx.x;
    const int b = batch[i];
    atomicAdd(&sums[(size_t)b * HDIM + j], h[(size_t)i * HDIM + j]);
    if (j == 0) atomicAdd(&counts[b], 1.0f);
}

// ---- head MLP: relu(concat(mean,sum) @ w1 + b1) @ w2 + b2, one block/graph ----
__global__ void head_kernel(const float* __restrict__ sums,
                            const float* __restrict__ counts,
                            const float* __restrict__ w1,
                            const float* __restrict__ b1,
                            const float* __restrict__ w2,
                            const float* __restrict__ b2,
                            float* __restrict__ out) {
    __shared__ float gv[2 * HDIM];
    __shared__ float red[HDIM];
    const int g = blockIdx.x;
    const int t = threadIdx.x;
    const float s = sums[(size_t)g * HDIM + t];
    const float c = fmaxf(counts[g], 1.0f);
    gv[t]        = s / c;
    gv[HDIM + t] = s;
    __syncthreads();
    float acc = b1[t];
#pragma unroll 8
    for (int i = 0; i < 2 * HDIM; ++i)
        acc = fmaf(gv[i], w1[(size_t)i * HDIM + t], acc);
    acc = fmaxf(acc, 0.0f);
    red[t] = acc * w2[t];
    __syncthreads();
    for (int off = HDIM / 2; off > 0; off >>= 1) {
        if (t < off) red[t] += red[t + off];
        __syncthreads();
    }
    if (t == 0) out[g] = red[0] + b2[0];
}

extern "C" void kernel_launch(void* const* d_in, const int* in_sizes, int n_in,
                              void* d_out, int out_size, void* d_ws, size_t ws_size,
                              hipStream_t stream) {
    const float* x      = (const float*)d_in[0];
    const int*   ei     = (const int*)d_in[1];
    const int    E      = in_sizes[1] / 2;
    const int*   src    = ei;
    const int*   dst    = ei + E;
    const int*   batch  = (const int*)d_in[2];
    const float* enc_w  = (const float*)d_in[3];
    const float* enc_b  = (const float*)d_in[4];
    const float* conv_w = (const float*)d_in[5];
    const float* conv_b = (const float*)d_in[6];
    const float* gam    = (const float*)d_in[7];
    const float* bet    = (const float*)d_in[8];
    const float* mu     = (const float*)d_in[9];
    const float* var    = (const float*)d_in[10];
    const float* w1     = (const float*)d_in[11];
    const float* b1     = (const float*)d_in[12];
    const float* w2     = (const float*)d_in[13];
    const float* b2     = (const float*)d_in[14];
    float*       out    = (float*)d_out;
    (void)n_in; (void)out_size; (void)ws_size;

    // workspace carve-up (~45 MB, L2-resident)
    char* ws = (char*)d_ws;
    size_t off = 0;
    auto alloc = [&](size_t bytes) -> void* {
        void* p = ws + off;
        off += (bytes + 255) & ~(size_t)255;
        return p;
    };
    float* h      = (float*)alloc((size_t)N_NODES * HDIM * sizeof(float));
    float* m      = (float*)alloc((size_t)N_NODES * HDIM * sizeof(float));
    float* dinv   = (float*)alloc((size_t)N_NODES * sizeof(float));
    int*   cnt    = (int*)  alloc((size_t)N_NODES * sizeof(int));
    int*   rowptr = (int*)  alloc((size_t)(N_NODES + 1) * sizeof(int));
    int*   cursor = (int*)  alloc((size_t)N_NODES * sizeof(int));
    int*   col    = (int*)  alloc((size_t)E * sizeof(int));
    float* wght   = (float*)alloc((size_t)E * sizeof(float));
    float* sums   = (float*)alloc((size_t)NGRAPH * HDIM * sizeof(float));
    float* cgs    = (float*)alloc((size_t)NGRAPH * sizeof(float));

    // ---- degrees, normalization, CSR-by-dst build (once per launch) ----
    hipMemsetAsync(cnt, 0, (size_t)N_NODES * sizeof(int), stream);
    deg_kernel<<<(E + 255) / 256, 256, 0, stream>>>(dst, cnt, E);
    dinv_kernel<<<(N_NODES + 255) / 256, 256, 0, stream>>>(cnt, dinv, N_NODES);
    scan_kernel<<<1, SCAN_T, 0, stream>>>(cnt, rowptr, N_NODES);
    hipMemcpyAsync(cursor, rowptr, (size_t)N_NODES * sizeof(int),
                   hipMemcpyDeviceToDevice, stream);
    fill_kernel<<<(E + 255) / 256, 256, 0, stream>>>(src, dst, dinv, cursor, col, wght, E);

    // ---- encoder GEMM: h = x @ enc_w + enc_b ----
    const int waves = (N_NODES / 16) * (HDIM / 16);   // 20000 waves
    wmma_gemm_kernel<NODE_DIMC, true><<<waves / 8, 256, 0, stream>>>(
        x, enc_w, enc_b, h, N_NODES, HDIM);

    // ---- GCN layers: WMMA transform + fused CSR aggregate/BN/residual ----
    const int aggBlocks = (N_NODES * 32 + 255) / 256;  // one wave per node
    for (int l = 0; l < LAYERS; ++l) {
        wmma_gemm_kernel<HDIM, false><<<waves / 8, 256, 0, stream>>>(
            h, conv_w + (size_t)l * HDIM * HDIM, nullptr, m, N_NODES, HDIM);
        aggregate_kernel<<<aggBlocks, 256, 0, stream>>>(
            m, rowptr, col, wght, dinv,
            conv_b + (size_t)l * HDIM, gam + (size_t)l * HDIM,
            bet + (size_t)l * HDIM, mu + (size_t)l * HDIM, var + (size_t)l * HDIM, h);
    }

    // ---- pooling + head ----
    hipMemsetAsync(sums, 0, (size_t)NGRAPH * HDIM * sizeof(float), stream);
    hipMemsetAsync(cgs,  0, (size_t)NGRAPH * sizeof(float), stream);
    pool_kernel<<<N_NODES, HDIM, 0, stream>>>(h, batch, sums, cgs);
    head_kernel<<<NGRAPH, HDIM, 0, stream>>>(sums, cgs, w1, b1, w2, b2, out);
}